// TensorDLT_13786845020560
// MI455X (gfx1250) — compile-verified
//
#include <hip/hip_runtime.h>

// ---------------------------------------------------------------------------
// TensorDLT for MI455X (gfx1250, wave32).
//   - memory-bound (~50 MB @ 23.3 TB/s => ~2.2us floor): one batch per lane,
//     b128 loads, single pass, coalesced output via LDS staging.
//   - Gram matrix M = A^T A (8x9 -> 9x9) computed per batch with two chained
//     V_WMMA_F32_16X16X4_F32 (K=4+4=8). Gram symmetry => A-operand == B-operand
//     registers (At[:,k:k+4](m,k)=A[k][m] matches A[k:k+4,:](k,n)=A[k][n]).
//   - Phase-2 is fully branch-free (clamped addresses + v_cndmask, per-lane
//     dump slots for the padded tile region) so EXEC stays all-ones across the
//     WMMAs (ISA requirement) and no saveexec/branch overhead per batch.
//   - vh[-1] of the economy SVD of an 8x9 matrix == eigenvector of the 2nd
//     smallest eigenvalue of M (smallest is exactly 0). Computed per lane with
//     ridged Cholesky + inverse iteration, deflating the exact null vector.
// ---------------------------------------------------------------------------

typedef float v8f __attribute__((ext_vector_type(8)));
typedef float v2f __attribute__((ext_vector_type(2)));

#define EPSF      1e-8f
#define NOISE_STD 1e-4f
#define SQRT2F    1.41421356237309515f

#define THREADS    256
#define REG_STRIDE 84   // dwords per batch region (>= 81 for M, float4 aligned)
#define DUMP_BASE  (THREADS * REG_STRIDE)   // 32 per-lane dump slots after regions

__device__ __forceinline__ constexpr int TRI(int i, int j) { return i * (i + 1) / 2 + j; } // i >= j

// Solve (L L^T) x = b given packed lower-tri Cholesky factor L and reciprocal
// diagonal invd. Fully unrolled -> stays in registers.
__device__ __forceinline__ void chol_solve(const float* L, const float* invd,
                                           const float* b, float* x) {
    float z[9];
#pragma unroll
    for (int i = 0; i < 9; ++i) {
        float s = b[i];
#pragma unroll
        for (int k = 0; k < i; ++k) s -= L[TRI(i, k)] * z[k];
        z[i] = s * invd[i];
    }
#pragma unroll
    for (int i = 8; i >= 0; --i) {
        float s = z[i];
#pragma unroll
        for (int k = i + 1; k < 9; ++k) s -= L[TRI(k, i)] * x[k];
        x[i] = s * invd[i];
    }
}

extern "C" __global__ void __launch_bounds__(THREADS)
dlt_wmma_kernel(const float* __restrict__ xA, const float* __restrict__ xB,
                const float* __restrict__ noise, float* __restrict__ out) {
    extern __shared__ float smem[];

    const int tid   = threadIdx.x;
    const int lane  = tid & 31;
    const int wave  = tid >> 5;
    const int rbase = tid * REG_STRIDE;
    const long batch = (long)blockIdx.x * THREADS + tid;

    // ------------------------- Phase 1: build A into LDS -------------------
    const float4 pA0 = ((const float4*)xA)[batch * 2 + 0];
    const float4 pA1 = ((const float4*)xA)[batch * 2 + 1];
    const float4 pB0 = ((const float4*)xB)[batch * 2 + 0];
    const float4 pB1 = ((const float4*)xB)[batch * 2 + 1];

    float xa[4] = {pA0.x, pA0.z, pA1.x, pA1.z};
    float ya[4] = {pA0.y, pA0.w, pA1.y, pA1.w};
    float xb[4] = {pB0.x, pB0.z, pB1.x, pB1.z};
    float yb[4] = {pB0.y, pB0.w, pB1.y, pB1.w};

    const float mxA = 0.25f * (xa[0] + xa[1] + xa[2] + xa[3]);
    const float myA = 0.25f * (ya[0] + ya[1] + ya[2] + ya[3]);
    const float mxB = 0.25f * (xb[0] + xb[1] + xb[2] + xb[3]);
    const float myB = 0.25f * (yb[0] + yb[1] + yb[2] + yb[3]);

    float dA = 0.f, dB = 0.f;
#pragma unroll
    for (int p = 0; p < 4; ++p) {
        float dx = xa[p] - mxA, dy = ya[p] - myA;
        dA += sqrtf(dx * dx + dy * dy);
        dx = xb[p] - mxB; dy = yb[p] - myB;
        dB += sqrtf(dx * dx + dy * dy);
    }
    const float sA = SQRT2F / (0.25f * dA + EPSF);
    const float sB = SQRT2F / (0.25f * dB + EPSF);

    float Areg[72];
#pragma unroll
    for (int p = 0; p < 4; ++p) {
        const float X1 = (xa[p] - mxA) * sA, Y1 = (ya[p] - myA) * sA;
        const float X2 = (xb[p] - mxB) * sB, Y2 = (yb[p] - myB) * sB;
        float* r0 = &Areg[(2 * p) * 9];
        float* r1 = &Areg[(2 * p + 1) * 9];
        r0[0] = 0.f;  r0[1] = 0.f;  r0[2] = 0.f;
        r0[3] = -X1;  r0[4] = -Y1;  r0[5] = -1.f;
        r0[6] = Y2 * X1; r0[7] = Y2 * Y1; r0[8] = Y2;
        r1[0] = X1;   r1[1] = Y1;   r1[2] = 1.f;
        r1[3] = 0.f;  r1[4] = 0.f;  r1[5] = 0.f;
        r1[6] = -X2 * X1; r1[7] = -X2 * Y1; r1[8] = -X2;
    }

    // noise is (B,8,9) contiguous: 72 floats == A laid out row-major -> fuse add
    const float4* nz = (const float4*)noise + batch * 18;
#pragma unroll
    for (int k = 0; k < 18; ++k) {
        const float4 nv = nz[k];
        const int i0 = 4 * k;
        float4 sv;
        sv.x = Areg[i0 + 0] + NOISE_STD * nv.x;
        sv.y = Areg[i0 + 1] + NOISE_STD * nv.y;
        sv.z = Areg[i0 + 2] + NOISE_STD * nv.z;
        sv.w = Areg[i0 + 3] + NOISE_STD * nv.w;
        *(float4*)&smem[rbase + i0] = sv;   // 16B aligned: 336*tid + 16*k
    }
    __syncthreads();

    // --------------- Phase 2: per-wave Gram matrices via WMMA --------------
    // A-operand 16x4 f32 layout: v0 = {K=0 | K=2}, v1 = {K=1 | K=3} across the
    // two half-waves; Gram symmetry lets the same registers feed SRC0 & SRC1.
    // Branch-free: loop-invariant clamped offsets + cndmask; EXEC stays full.
    {
        const int  col    = lane & 15;
        const int  half   = lane >> 4;
        const bool colOK  = (col < 9);
        const int  cc     = colOK ? col : 8;            // clamped in-region col
        const int  l0     = (2 * half + 0) * 9 + cc;
        const int  l1     = (2 * half + 1) * 9 + cc;
        const int  l2     = (2 * half + 4) * 9 + cc;
        const int  l3     = (2 * half + 5) * 9 + cc;
        const int  dumpIx = DUMP_BASE + lane;           // private dump slot

        bool sok[8];
        int  srel[8];
#pragma unroll
        for (int d = 0; d < 8; ++d) {
            const int row = d + 8 * half;
            sok[d]  = (row < 9) && colOK;
            srel[d] = row * 9 + col;
        }

#pragma unroll 2
        for (int jj = 0; jj < 32; ++jj) {
            const int base = ((wave << 5) + jj) * REG_STRIDE;
            const float t0 = smem[base + l0];
            const float t1 = smem[base + l1];
            const float t2 = smem[base + l2];
            const float t3 = smem[base + l3];
            v2f a0, a1;
            a0[0] = colOK ? t0 : 0.f;   // zero-pad tile cols 9..15
            a0[1] = colOK ? t1 : 0.f;
            a1[0] = colOK ? t2 : 0.f;
            a1[1] = colOK ? t3 : 0.f;

            v8f c = {0.f, 0.f, 0.f, 0.f, 0.f, 0.f, 0.f, 0.f};
            c = __builtin_amdgcn_wmma_f32_16x16x4_f32(false, a0, false, a0,
                                                      (short)0, c, false, false);
            c = __builtin_amdgcn_wmma_f32_16x16x4_f32(false, a1, false, a1,
                                                      (short)0, c, false, false);

            // Store the 9x9 block of M over the dead A region; invalid lanes
            // write to their private dump slot (no divergence).
            // C layout: VGPR d, lanes 0-15 -> row d, lanes 16-31 -> row d+8.
#pragma unroll
            for (int d = 0; d < 8; ++d) {
                const int addr = sok[d] ? (base + srel[d]) : dumpIx;
                smem[addr] = c[d];
            }
        }
    }
    __syncthreads();

    // ------- Phase 3: per-lane eigen-solve (vh[-1]) + homography compose ----
    float m[45];
#pragma unroll
    for (int i = 0; i < 9; ++i)
#pragma unroll
        for (int j = 0; j <= i; ++j)
            m[TRI(i, j)] = smem[rbase + i * 9 + j];

    float tr = 0.f;
#pragma unroll
    for (int i = 0; i < 9; ++i) tr += m[TRI(i, i)];
    const float ridge = 1e-6f * tr + 1e-12f;
#pragma unroll
    for (int i = 0; i < 9; ++i) m[TRI(i, i)] += ridge;

    // In-place Cholesky, reciprocal diagonal kept separately.
    float invd[9];
#pragma unroll
    for (int j = 0; j < 9; ++j) {
        float d = m[TRI(j, j)];
#pragma unroll
        for (int k = 0; k < j; ++k) { const float l = m[TRI(j, k)]; d -= l * l; }
        d = sqrtf(fmaxf(d, 1e-24f));
        const float inv = 1.0f / d;
        m[TRI(j, j)] = d;
        invd[j] = inv;
#pragma unroll
        for (int i = j + 1; i < 9; ++i) {
            float s = m[TRI(i, j)];
#pragma unroll
            for (int k = 0; k < j; ++k) s -= m[TRI(i, k)] * m[TRI(j, k)];
            m[TRI(i, j)] = s * inv;
        }
    }

    // Null vector n (sigma9 == 0): inverse iteration on M + ridge*I.
    float nv[9];
#pragma unroll
    for (int i = 0; i < 9; ++i) nv[i] = 1.0f;
#pragma unroll
    for (int t = 0; t < 2; ++t) {
        float z[9];
        chol_solve(m, invd, nv, z);
        float nn = 0.f;
#pragma unroll
        for (int i = 0; i < 9; ++i) nn += z[i] * z[i];
        const float s = rsqrtf(nn + 1e-30f);
#pragma unroll
        for (int i = 0; i < 9; ++i) nv[i] = z[i] * s;
    }

    // vh[-1]: deflated inverse iteration (project out n each step) -> sigma8 vec.
    float y[9];
#pragma unroll
    for (int i = 0; i < 9; ++i) y[i] = (i & 1) ? -1.0f : 1.0f;
    {
        float d0 = 0.f;
#pragma unroll
        for (int i = 0; i < 9; ++i) d0 += y[i] * nv[i];
#pragma unroll
        for (int i = 0; i < 9; ++i) y[i] -= d0 * nv[i];
    }
#pragma unroll
    for (int t = 0; t < 3; ++t) {
        float z[9];
        chol_solve(m, invd, y, z);
        float dot = 0.f;
#pragma unroll
        for (int i = 0; i < 9; ++i) dot += z[i] * nv[i];
#pragma unroll
        for (int i = 0; i < 9; ++i) z[i] -= dot * nv[i];
        float nn = 0.f;
#pragma unroll
        for (int i = 0; i < 9; ++i) nn += z[i] * z[i];
        const float s = rsqrtf(nn + 1e-30f);
#pragma unroll
        for (int i = 0; i < 9; ++i) y[i] = z[i] * s;
    }

    // H = inv(TB) @ (Hn @ TA), then normalize by H[2,2] + eps.
    float G[9];
#pragma unroll
    for (int i = 0; i < 3; ++i) {
        G[i * 3 + 0] = y[i * 3 + 0] * sA;
        G[i * 3 + 1] = y[i * 3 + 1] * sA;
        G[i * 3 + 2] = -sA * mxA * y[i * 3 + 0] - sA * myA * y[i * 3 + 1] + y[i * 3 + 2];
    }
    const float invsB = 1.0f / sB;
    float Hm[9];
#pragma unroll
    for (int j = 0; j < 3; ++j) {
        Hm[0 * 3 + j] = invsB * G[0 * 3 + j] + mxB * G[2 * 3 + j];
        Hm[1 * 3 + j] = invsB * G[1 * 3 + j] + myB * G[2 * 3 + j];
        Hm[2 * 3 + j] = G[2 * 3 + j];
    }
    const float w = 1.0f / (Hm[8] + EPSF);
#pragma unroll
    for (int k = 0; k < 9; ++k) smem[rbase + k] = Hm[k] * w;
    __syncthreads();

    // ------------------ Phase 4: coalesced output store --------------------
    const long outBase = (long)blockIdx.x * THREADS * 9;
#pragma unroll
    for (int it = 0; it < 9; ++it) {
        const unsigned g = (unsigned)tid + THREADS * it;   // 0 .. 2303
        const unsigned j = g / 9u;
        const unsigned k = g - j * 9u;
        out[outBase + g] = smem[j * REG_STRIDE + k];
    }
}

extern "C" void kernel_launch(void* const* d_in, const int* in_sizes, int n_in,
                              void* d_out, int out_size, void* d_ws, size_t ws_size,
                              hipStream_t stream) {
    (void)n_in; (void)out_size; (void)d_ws; (void)ws_size;
    const float* xA    = (const float*)d_in[0];
    const float* xB    = (const float*)d_in[1];
    const float* noise = (const float*)d_in[2];
    float* out = (float*)d_out;

    const int B = in_sizes[0] / 8;                  // xA: (B,4,2)
    const int blocks = (B + THREADS - 1) / THREADS; // B = 131072 -> 512 blocks
    const size_t shmem = (size_t)(THREADS * REG_STRIDE + 32) * sizeof(float); // ~84 KB

    dlt_wmma_kernel<<<blocks, THREADS, shmem, stream>>>(xA, xB, noise, out);
}